// LANT_18588618457551
// MI455X (gfx1250) — compile-verified
//
#include <hip/hip_runtime.h>
#include <hip/hip_bf16.h>

// GAT + PReLU for MI455X (gfx1250, wave32, WMMA).
// N=50000 nodes, F=128 in, HEADS=2, HID=64 (HEADS*HID=128 out), E=800000 edges.

#define F_IN   128
#define F_OUT  128
#define HEADS  2
#define HID    64
#define NEG_SLOPE 0.2f

typedef __attribute__((ext_vector_type(2))) float v2f;
typedef __attribute__((ext_vector_type(8))) float v8f;

// ---- order-preserving float<->uint mapping for atomic max over signed floats
__device__ __forceinline__ unsigned f2ord(float f) {
    unsigned u = __float_as_uint(f);
    return (u & 0x80000000u) ? ~u : (u | 0x80000000u);
}
__device__ __forceinline__ float ord2f(unsigned u) {
    return (u & 0x80000000u) ? __uint_as_float(u & 0x7FFFFFFFu)
                             : __uint_as_float(~u);
}
#define ORD_NEG_INF 0x007FFFFFu  // f2ord(-inf)

__device__ __forceinline__ float lrelu(float v) {
    return v >= 0.f ? v : NEG_SLOPE * v;
}

// ---------------------------------------------------------------------------
// 0) init: zero output accumulator + denom, set running-max to -inf
__global__ void init_k(float* __restrict__ out, float* __restrict__ denom,
                       unsigned* __restrict__ mmax, int N) {
    int i = blockIdx.x * blockDim.x + threadIdx.x;
    if (i < N * F_OUT) out[i] = 0.f;
    if (i < N * HEADS) { denom[i] = 0.f; mmax[i] = ORD_NEG_INF; }
}

// ---------------------------------------------------------------------------
// 1) fold attention vectors through W:  w_eff[k][h] = sum_c W[k][h*HID+c]*att[h][c]
//    (turns the second full GEMM of the reference into a 128x2 matvec)
__global__ void prep_w_k(const float* __restrict__ W_src, const float* __restrict__ W_dst,
                         const float* __restrict__ att_src, const float* __restrict__ att_dst,
                         float* __restrict__ wse, float* __restrict__ wde) {
    int t = threadIdx.x;           // 256 threads: (k,h) pairs
    int k = t & (F_IN - 1);
    int h = t >> 7;
    float s = 0.f, d = 0.f;
    #pragma unroll 8
    for (int c = 0; c < HID; ++c) {
        s += W_src[k * F_OUT + h * HID + c] * att_src[h * HID + c];
        d += W_dst[k * F_OUT + h * HID + c] * att_dst[h * HID + c];
    }
    wse[k * HEADS + h] = s;
    wde[k * HEADS + h] = d;
}

// ---------------------------------------------------------------------------
// 2) xs = x @ W_src via V_WMMA_F32_16X16X4_F32 (full fp32 precision).
//    One wave per 16x16 D-tile; 8 waves/block share one M-row of tiles.
__global__ void gemm_xs_k(const float* __restrict__ x, const float* __restrict__ W,
                          float* __restrict__ xs, int Mtiles, int N) {
    const int wid  = (blockIdx.x * blockDim.x + threadIdx.x) >> 5; // wave-uniform
    const int lane = threadIdx.x & 31;
    const int tile_m = wid >> 3;   // 8 N-tiles per M-row (F_OUT/16 = 8)
    const int tile_n = wid & 7;
    if (tile_m >= Mtiles) return;  // uniform across wave -> EXEC all-1s inside

    const int l    = lane & 15;
    const int half = lane >> 4;

    const int m0 = tile_m * 16;
    const int n0 = tile_n * 16;
    const bool full = (m0 + 16 <= N);          // wave-uniform fast path
    const int arow_idx = full ? (m0 + l) : min(m0 + l, N - 1);

    // A-frag: row M = lane%16; lanes 0-15 carry K={k,k+1}, lanes 16-31 K={k+2,k+3}
    const float* arow = x + (size_t)arow_idx * F_IN + 2 * half;
    // B-frag: col N = n0 + lane%16; lanes 0-15 rows K={k,k+1}, lanes 16-31 K={k+2,k+3}
    const float* bcol = W + (size_t)(2 * half) * F_OUT + n0 + l;

    v8f acc = {0.f, 0.f, 0.f, 0.f, 0.f, 0.f, 0.f, 0.f};
    #pragma unroll 4
    for (int k = 0; k < F_IN; k += 4) {
        v2f a = *(const v2f*)(arow + k);                 // global_load_b64
        v2f b;
        b.x = bcol[(size_t)k * F_OUT];
        b.y = bcol[(size_t)k * F_OUT + F_OUT];
        acc = __builtin_amdgcn_wmma_f32_16x16x4_f32(
            /*neg_a=*/false, a, /*neg_b=*/false, b,
            /*c_mod=*/(short)0, acc, /*reuse_a=*/false, /*reuse_b=*/false);
    }

    // D layout: VGPR r holds row (m0 + r + 8*half), col n0 + lane%16
    float* orow = xs + (size_t)(m0 + 8 * half) * F_OUT + n0 + l;
    if (full) {
        // straight-line stores, no exec-mask churn (always taken: N % 16 == 0)
        #pragma unroll
        for (int r = 0; r < 8; ++r) orow[(size_t)r * F_OUT] = acc[r];
    } else {
        #pragma unroll
        for (int r = 0; r < 8; ++r) {
            if (m0 + 8 * half + r < N) orow[(size_t)r * F_OUT] = acc[r];
        }
    }
}

// ---------------------------------------------------------------------------
// 3) per-node attention logits: a_src[n][h] = x[n,:]·w_eff[:,h]  (wave per node)
__global__ void compute_a_k(const float* __restrict__ x,
                            const float* __restrict__ wse, const float* __restrict__ wde,
                            float* __restrict__ a_src, float* __restrict__ a_dst, int N) {
    const int n    = (blockIdx.x * blockDim.x + threadIdx.x) >> 5;
    const int lane = threadIdx.x & 31;
    if (n >= N) return;
    float4 xv = ((const float4*)(x + (size_t)n * F_IN))[lane];
    const int k0 = lane * 4;
    float ps0, ps1, pd0, pd1;
    ps0 = xv.x * wse[(k0+0)*2+0] + xv.y * wse[(k0+1)*2+0] + xv.z * wse[(k0+2)*2+0] + xv.w * wse[(k0+3)*2+0];
    ps1 = xv.x * wse[(k0+0)*2+1] + xv.y * wse[(k0+1)*2+1] + xv.z * wse[(k0+2)*2+1] + xv.w * wse[(k0+3)*2+1];
    pd0 = xv.x * wde[(k0+0)*2+0] + xv.y * wde[(k0+1)*2+0] + xv.z * wde[(k0+2)*2+0] + xv.w * wde[(k0+3)*2+0];
    pd1 = xv.x * wde[(k0+0)*2+1] + xv.y * wde[(k0+1)*2+1] + xv.z * wde[(k0+2)*2+1] + xv.w * wde[(k0+3)*2+1];
    #pragma unroll
    for (int off = 16; off > 0; off >>= 1) {
        ps0 += __shfl_down(ps0, off, 32);
        ps1 += __shfl_down(ps1, off, 32);
        pd0 += __shfl_down(pd0, off, 32);
        pd1 += __shfl_down(pd1, off, 32);
    }
    if (lane == 0) {
        a_src[n * 2 + 0] = ps0;  a_src[n * 2 + 1] = ps1;
        a_dst[n * 2 + 0] = pd0;  a_dst[n * 2 + 1] = pd1;
    }
}

// ---------------------------------------------------------------------------
// 4) segment max of leaky_relu(a_src[src]+a_dst[dst]) over incoming edges
__global__ void edge_max_k(const int* __restrict__ ei,
                           const float* __restrict__ a_src, const float* __restrict__ a_dst,
                           unsigned* __restrict__ mmax, int E) {
    int e = blockIdx.x * blockDim.x + threadIdx.x;
    if (e >= E) return;
    int s = ei[e], d = ei[E + e];
    #pragma unroll
    for (int h = 0; h < HEADS; ++h) {
        float v = lrelu(a_src[s * 2 + h] + a_dst[d * 2 + h]);
        atomicMax(&mmax[d * 2 + h], f2ord(v));
    }
}

// ---------------------------------------------------------------------------
// 5) ee = exp(e - max); denom = segment_sum(ee)
__global__ void edge_exp_k(const int* __restrict__ ei,
                           const float* __restrict__ a_src, const float* __restrict__ a_dst,
                           const unsigned* __restrict__ mmax,
                           float* __restrict__ ee, float* __restrict__ denom, int E) {
    int e = blockIdx.x * blockDim.x + threadIdx.x;
    if (e >= E) return;
    int s = ei[e], d = ei[E + e];
    #pragma unroll
    for (int h = 0; h < HEADS; ++h) {
        float v = lrelu(a_src[s * 2 + h] + a_dst[d * 2 + h]);
        float m = ord2f(mmax[d * 2 + h]);
        float x = __expf(v - m);
        ee[(size_t)e * 2 + h] = x;
        atomicAdd(&denom[d * 2 + h], x);
    }
}

// ---------------------------------------------------------------------------
// 6) scatter messages: out[dst] += xs[src] * alpha   (128 lanes per edge,
//    coalesced 512B gather of xs row + L2-resident f32 atomics)
__global__ void edge_scatter_k(const int* __restrict__ ei,
                               const float* __restrict__ xs, const float* __restrict__ ee,
                               const float* __restrict__ denom,
                               float* __restrict__ out, int E) {
    int gid = blockIdx.x * blockDim.x + threadIdx.x;   // E*128 < 2^31
    int e = gid >> 7;
    int c = gid & (F_OUT - 1);
    if (e >= E) return;
    int s = ei[e], d = ei[E + e];
    int h = c >> 6;                                    // c/HID
    float alpha = ee[(size_t)e * 2 + h] / (denom[d * 2 + h] + 1e-16f);
    atomicAdd(&out[(size_t)d * F_OUT + c], xs[(size_t)s * F_OUT + c] * alpha);
}

// ---------------------------------------------------------------------------
// 7) bias + PReLU, in place on the accumulator (= d_out)
__global__ void finalize_k(float* __restrict__ out, const float* __restrict__ bias,
                           const float* __restrict__ prelu_w, int N) {
    int i = blockIdx.x * blockDim.x + threadIdx.x;
    if (i >= N * F_OUT) return;
    int c = i & (F_OUT - 1);
    float v = out[i] + bias[c];
    out[i] = v >= 0.f ? v : prelu_w[c] * v;
}

// ---------------------------------------------------------------------------
extern "C" void kernel_launch(void* const* d_in, const int* in_sizes, int n_in,
                              void* d_out, int out_size, void* d_ws, size_t ws_size,
                              hipStream_t stream) {
    const float* x       = (const float*)d_in[0];
    const float* W_src   = (const float*)d_in[1];
    const float* W_dst   = (const float*)d_in[2];
    const float* att_src = (const float*)d_in[3];
    const float* att_dst = (const float*)d_in[4];
    const float* bias    = (const float*)d_in[5];
    const float* prelu_w = (const float*)d_in[6];
    const int*   ei      = (const int*)d_in[7];
    float* out = (float*)d_out;

    const int N = in_sizes[0] / F_IN;   // 50000
    const int E = in_sizes[7] / 2;      // 800000

    // workspace layout (floats)
    float*    xs    = (float*)d_ws;                 // N*128
    float*    a_src = xs + (size_t)N * F_OUT;       // N*2
    float*    a_dst = a_src + (size_t)N * HEADS;    // N*2
    unsigned* mmax  = (unsigned*)(a_dst + (size_t)N * HEADS); // N*2
    float*    denom = (float*)(mmax + (size_t)N * HEADS);     // N*2
    float*    ee    = denom + (size_t)N * HEADS;    // E*2
    float*    wse   = ee + (size_t)E * HEADS;       // 128*2
    float*    wde   = wse + F_IN * HEADS;           // 128*2

    const int B = 256;
    const int Mtiles = (N + 15) / 16;

    init_k<<<(N * F_OUT + B - 1) / B, B, 0, stream>>>(out, denom, mmax, N);
    prep_w_k<<<1, B, 0, stream>>>(W_src, W_dst, att_src, att_dst, wse, wde);
    gemm_xs_k<<<Mtiles, B, 0, stream>>>(x, W_src, xs, Mtiles, N);   // 8 waves = 8 N-tiles
    compute_a_k<<<(N * 32 + B - 1) / B, B, 0, stream>>>(x, wse, wde, a_src, a_dst, N);
    edge_max_k<<<(E + B - 1) / B, B, 0, stream>>>(ei, a_src, a_dst, mmax, E);
    edge_exp_k<<<(E + B - 1) / B, B, 0, stream>>>(ei, a_src, a_dst, mmax, ee, denom, E);
    edge_scatter_k<<<(E * F_OUT + B - 1) / B, B, 0, stream>>>(ei, xs, ee, denom, out, E);
    finalize_k<<<(N * F_OUT + B - 1) / B, B, 0, stream>>>(out, bias, prelu_w, N);
}